// OrderingOperator_34883724378527
// MI455X (gfx1250) — compile-verified
//
#include <hip/hip_runtime.h>

// O = (L*X - colsum(X)) / (L-1), X: [B=8, L=4096, D=1024] fp32.
// Bandwidth-bound (AI ~0.6 FLOP/B). Two-pass; X (128MiB) stays resident in the
// 192MB L2 between passes. Column reduction runs on the matrix pipes via
// V_WMMA_F32_16X16X4_F32 with an all-ones A operand (D[m][n] = sum_k B[k][n] + C),
// which is exact in fp32.

typedef __attribute__((ext_vector_type(2))) float v2f;
typedef __attribute__((ext_vector_type(4))) float v4f;
typedef __attribute__((ext_vector_type(8))) float v8f;

constexpr int    BB   = 8;
constexpr int    LL   = 4096;
constexpr int    DD   = 1024;
constexpr int    RB   = 8;          // row blocks per batch
constexpr int    ROWS = LL / RB;    // 512 rows per block
constexpr int    CT   = 8;          // column tiles of 128 (8 waves * 16 cols)
constexpr float  INVL1 = 1.0f / (float)(LL - 1);

// ---------------------------------------------------------------------------
// Pass 1: per-(batch, row-block) column sums via WMMA f32 16x16x4.
// grid = B*CT*RB = 512 blocks, 256 threads (8 waves). Wave w owns 16 columns.
// Each v_wmma consumes a 4x16 tile of X (B operand); A = ones => column sums
// accumulate into the f32 C operand. Two accumulators break the RAW chain.
// ---------------------------------------------------------------------------
__global__ __launch_bounds__(256) void colsum_wmma(const float* __restrict__ X,
                                                   float* __restrict__ partial) {
  const int bid   = blockIdx.x;
  const int b     = bid >> 6;        // 64 blocks per batch (8 ctiles * 8 rblocks)
  const int rem   = bid & 63;
  const int ctile = rem >> 3;
  const int rb    = rem & 7;
  const int wave  = threadIdx.x >> 5;
  const int lane  = threadIdx.x & 31;
  const int half  = lane >> 4;       // 0: K=0,1 rows; 1: K=2,3 rows
  const int n     = lane & 15;       // column within the wave's 16-col strip
  const int col   = ctile * 128 + wave * 16 + n;

  // element index of X[b, rb*ROWS + 2*half, col]
  size_t p = ((size_t)b * LL + (size_t)rb * ROWS + 2 * half) * DD + col;

  const v2f ones = {1.0f, 1.0f};
  v8f acc0 = {};
  v8f acc1 = {};

  // 512 rows, 8 rows (= 2 WMMAs) per iteration.
#pragma unroll 4
  for (int it = 0; it < ROWS / 8; ++it) {
    v2f b0, b1;
    b0.x = X[p];
    b0.y = X[p + DD];
    b1.x = X[p + 4 * (size_t)DD];
    b1.y = X[p + 5 * (size_t)DD];
    acc0 = __builtin_amdgcn_wmma_f32_16x16x4_f32(false, ones, false, b0,
                                                 (short)0, acc0, false, false);
    acc1 = __builtin_amdgcn_wmma_f32_16x16x4_f32(false, ones, false, b1,
                                                 (short)0, acc1, false, false);
    p += 8 * (size_t)DD;
  }

  // All D rows are identical (A was ones): lane l's acc[0] = colsum(col l&15).
  const float s = acc0[0] + acc1[0];
  if (lane < 16) {
    partial[((size_t)b * RB + rb) * DD + (size_t)ctile * 128 + wave * 16 + n] = s;
  }
}

// ---------------------------------------------------------------------------
// Pass 1b: fold the RB row-block partials into totals[b][d]. 8192 threads.
// Fixed-order sum -> deterministic.
// ---------------------------------------------------------------------------
__global__ __launch_bounds__(256) void finish_total(const float* __restrict__ partial,
                                                    float* __restrict__ totals) {
  const int t = blockIdx.x * 256 + threadIdx.x;   // [0, B*D)
  const int b = t >> 10;
  const int d = t & 1023;
  float s = 0.0f;
#pragma unroll
  for (int rb = 0; rb < RB; ++rb) {
    s += partial[((size_t)b * RB + rb) * DD + d];
  }
  totals[t] = s;
}

// ---------------------------------------------------------------------------
// Pass 2: O = (L*x - total) * (1/(L-1)), float4-vectorized.
// X reads hit L2 (X fits in the 192MB L2); mark them non-temporal (dead after
// read) and stream the output with non-temporal stores so the 128MiB of O
// doesn't evict not-yet-read X lines.
// ---------------------------------------------------------------------------
__global__ __launch_bounds__(256) void apply_k(const v4f* __restrict__ X4,
                                               const v4f* __restrict__ T4,
                                               v4f* __restrict__ O4) {
  const size_t i   = (size_t)blockIdx.x * 256 + threadIdx.x;  // float4 index
  const int    dv  = (int)(i & 255);        // 256 float4 per row of D=1024
  const size_t row = i >> 8;                // b*L + r
  const int    b   = (int)(row >> 12);      // L = 4096 rows per batch

  const v4f x = __builtin_nontemporal_load(&X4[i]);
  const v4f t = T4[b * 256 + dv];
  const v4f o = ((float)LL * x - t) * INVL1;
  __builtin_nontemporal_store(o, &O4[i]);
}

// ---------------------------------------------------------------------------
extern "C" void kernel_launch(void* const* d_in, const int* in_sizes, int n_in,
                              void* d_out, int out_size, void* d_ws, size_t ws_size,
                              hipStream_t stream) {
  (void)in_sizes; (void)n_in; (void)out_size; (void)ws_size;

  const float* X   = (const float*)d_in[0];
  float*       out = (float*)d_out;

  // Workspace layout: partials [B][RB][D] (256 KiB) then totals [B][D] (32 KiB).
  float* partial = (float*)d_ws;
  float* totals  = partial + (size_t)BB * RB * DD;

  colsum_wmma<<<BB * CT * RB, 256, 0, stream>>>(X, partial);
  finish_total<<<(BB * DD) / 256, 256, 0, stream>>>(partial, totals);

  const int n4 = BB * LL * DD / 4;                 // 8,388,608 float4
  apply_k<<<n4 / 256, 256, 0, stream>>>((const v4f*)X, (const v4f*)totals,
                                        (v4f*)out);
}